// ConnectionsSMPV3_83760452206686
// MI455X (gfx1250) — compile-verified
//
#include <hip/hip_runtime.h>
#include <math.h>

#define SEGS 65
#define CCH  64
#define PIX  (256*256)
#define BB   4
#define SDIM 64

typedef __attribute__((ext_vector_type(2))) float v2f;
typedef __attribute__((ext_vector_type(8))) float v8f;

// Monotone order-preserving float <-> uint transform: max over transformed
// uints == max over floats. Lets us use a single non-divergent ds_max_u32 /
// global_atomic_max_u32 per element instead of a two-path signed trick.
__device__ __forceinline__ unsigned ftos(float f) {
    unsigned b = __float_as_uint(f);
    return (b & 0x80000000u) ? ~b : (b | 0x80000000u);
}
__device__ __forceinline__ float stof(unsigned u) {
    unsigned b = (u & 0x80000000u) ? (u & 0x7FFFFFFFu) : ~u;
    return __uint_as_float(b);
}
#define NEG_INF_S 0x007FFFFFu   // ftos(-inf)

__global__ void seg_init_kernel(unsigned* seg, int n) {
    int i = blockIdx.x * blockDim.x + threadIdx.x;
    if (i < n) seg[i] = NEG_INF_S;
}

// LDS-staged segment max. Each block reduces 2048 pixels x 64 channels into a
// 65x64 LDS table (transformed-uint space), then flushes with global u32-max
// atomics. float4/int4 loads -> global_load_b128 (512 B per wave per issue).
__global__ void segmax_kernel(const float* __restrict__ enc,
                              const int*   __restrict__ masks,
                              unsigned*    __restrict__ seg) {
    const int pixPerBlock    = 2048;
    const int chunksPerBatch = PIX / pixPerBlock;   // 32
    int b     = blockIdx.x / chunksPerBatch;
    int chunk = blockIdx.x % chunksPerBatch;
    int p0    = chunk * pixPerBlock;

    __shared__ unsigned tab[SEGS * CCH];            // 16.6 KB
    for (int i = threadIdx.x; i < SEGS * CCH; i += blockDim.x) tab[i] = NEG_INF_S;
    __syncthreads();

    const float* encb  = enc   + (size_t)b * CCH * PIX;
    const int*   maskb = masks + (size_t)b * PIX;

    for (int base = 0; base < pixPerBlock; base += blockDim.x * 4) {
        int p = p0 + base + threadIdx.x * 4;        // 16B-aligned
        int4 id4 = *(const int4*)&maskb[p];
        unsigned* r0 = &tab[id4.x * CCH];
        unsigned* r1 = &tab[id4.y * CCH];
        unsigned* r2 = &tab[id4.z * CCH];
        unsigned* r3 = &tab[id4.w * CCH];
        #pragma unroll 4
        for (int c = 0; c < CCH; ++c) {             // b128 coalesced across lanes
            float4 v = *(const float4*)&encb[(size_t)c * PIX + p];
            atomicMax(&r0[c], ftos(v.x));
            atomicMax(&r1[c], ftos(v.y));
            atomicMax(&r2[c], ftos(v.z));
            atomicMax(&r3[c], ftos(v.w));
        }
    }
    __syncthreads();

    unsigned* segb = seg + (size_t)b * SEGS * CCH;
    for (int i = threadIdx.x; i < SEGS * CCH; i += blockDim.x)
        atomicMax(&segb[i], tab[i]);
}

// vec[b,c,s] = seg[b, s+1, c]   (first output, (B,C,64,1) flat)
__global__ void vec_out_kernel(const unsigned* __restrict__ seg,
                               float* __restrict__ outv) {
    int i = blockIdx.x * blockDim.x + threadIdx.x;
    if (i >= BB * CCH * SDIM) return;
    int s = i % SDIM;
    int c = (i / SDIM) % CCH;
    int b = i / (SDIM * CCH);
    outv[i] = stof(seg[((size_t)b * SEGS + (s + 1)) * CCH + c]);
}

// Fused outer-product + 3x3 conv + bias + sigmoid using V_WMMA_F32_16X16X4_F32.
// One wave computes a 16-position x 4-outchannel tile; K = 9 taps * 64 ch = 576.
__global__ void conv_wmma_kernel(const unsigned* __restrict__ seg,
                                 const float* __restrict__ w,
                                 const float* __restrict__ bias,
                                 float*       __restrict__ outc) {
    __shared__ float vpad[CCH * 66];   // v padded with a zero column on each side
    __shared__ float wl[4 * 576];      // wl[o][t][c] = w[o][c][t]

    int b        = blockIdx.x >> 5;            // 32 blocks per batch
    int tileBase = (blockIdx.x & 31) * 8;      // 8 waves per block
    const unsigned* segb = seg + (size_t)b * SEGS * CCH;

    for (int i = threadIdx.x; i < CCH * 66; i += blockDim.x) {
        int c = i / 66, x = i % 66;
        vpad[i] = (x == 0 || x == 65) ? 0.0f : stof(segb[x * CCH + c]);
    }
    for (int i = threadIdx.x; i < 4 * 576; i += blockDim.x) {
        int o = i / 576, r = i % 576;
        int t = r / 64, c = r % 64;
        wl[i] = w[((size_t)(o * CCH + c)) * 9 + t];
    }
    __syncthreads();

    int wave = threadIdx.x >> 5;
    int lane = threadIdx.x & 31;
    int m    = lane & 15;          // A row (spatial j-offset) AND B column (out channel)
    int half = lane >> 4;
    int kb   = half * 2;           // K pair within each 4-chunk per ISA f32 layout

    int tile = tileBase + wave;    // [0,256)
    int ic   = tile >> 2;          // spatial row i
    int j0   = (tile & 3) << 4;    // spatial col base

    bool wvalid = (m < 4);
    int  mo     = wvalid ? m : 0;  // clamp so LDS reads stay in-bounds

    v8f acc = {0.f, 0.f, 0.f, 0.f, 0.f, 0.f, 0.f, 0.f};

    for (int t = 0; t < 9; ++t) {
        int di = t / 3 - 1, dj = t % 3 - 1;
        int xi = ic + di + 1;            // [0,65] into padded table
        int xj = j0 + m + dj + 1;        // [0,65]
        const float* wrow = &wl[mo * 576 + t * 64];
        #pragma unroll
        for (int c4 = 0; c4 < 16; ++c4) {
            int c0 = c4 * 4 + kb;
            v2f a, bv;
            a.x = vpad[c0 * 66 + xi] * vpad[c0 * 66 + xj];           // vmat im2col
            a.y = vpad[(c0 + 1) * 66 + xi] * vpad[(c0 + 1) * 66 + xj];
            bv.x = wvalid ? wrow[c0]     : 0.0f;
            bv.y = wvalid ? wrow[c0 + 1] : 0.0f;
            acc = __builtin_amdgcn_wmma_f32_16x16x4_f32(
                      false, a, false, bv, (short)0, acc, false, false);
        }
    }

    if (m < 4) {                         // lanes whose column is a real out channel
        float bs = bias[m];
        #pragma unroll
        for (int r = 0; r < 8; ++r) {
            int j = j0 + r + half * 8;   // D layout: VGPR r -> M=r (+8 for upper half)
            float val = acc[r] + bs;
            outc[(((size_t)b * 4 + m) * SDIM + ic) * SDIM + j] =
                1.0f / (1.0f + expf(-val));
        }
    }
}

extern "C" void kernel_launch(void* const* d_in, const int* in_sizes, int n_in,
                              void* d_out, int out_size, void* d_ws, size_t ws_size,
                              hipStream_t stream) {
    const float* enc   = (const float*)d_in[0];
    const int*   masks = (const int*)d_in[1];
    const float* w     = (const float*)d_in[2];
    const float* bias  = (const float*)d_in[3];
    float*    out  = (float*)d_out;
    unsigned* seg  = (unsigned*)d_ws;           // B*65*64 u32 = 66,560 B
    float*    outv = out;                       // 16384 floats (vec)
    float*    outc = out + BB * CCH * SDIM;     // 65536 floats (connections)

    int nseg = BB * SEGS * CCH;
    seg_init_kernel<<<(nseg + 255) / 256, 256, 0, stream>>>(seg, nseg);
    segmax_kernel<<<BB * 32, 256, 0, stream>>>(enc, masks, seg);
    vec_out_kernel<<<(BB * CCH * SDIM + 255) / 256, 256, 0, stream>>>(seg, outv);
    conv_wmma_kernel<<<BB * 32, 256, 0, stream>>>(seg, w, bias, outc);
}